// NGCF_6021544149547
// MI455X (gfx1250) — compile-verified
//
#include <hip/hip_runtime.h>

#define NUM_USERS 100000
#define N_NODES   300000
#define DIM       64
#define TILE      16
#define OUT_COLS  256  // (3 layers + 1) * 64

typedef __attribute__((ext_vector_type(2))) float v2f;
typedef __attribute__((ext_vector_type(4))) float v4f;
typedef __attribute__((ext_vector_type(8))) float v8f;

// ---------------------------------------------------------------------------
// Kernel 1: build all_emb = concat(user_emb, item_emb), zero side accumulator,
// and write the un-normalized layer-0 embedding into out[:, 0:64].
// Vectorized: one float4 per thread (rows are 16 float4s; the user/item
// boundary is row-aligned so a float4 never straddles it).
// ---------------------------------------------------------------------------
__global__ void __launch_bounds__(256) ngcf_init(
    const float* __restrict__ user_emb, const float* __restrict__ item_emb,
    float* __restrict__ emb, float* __restrict__ side, float* __restrict__ out)
{
    long i4 = (long)blockIdx.x * blockDim.x + threadIdx.x;   // over N_NODES*16
    if (i4 >= (long)N_NODES * (DIM / 4)) return;
    long n  = i4 >> 4;                 // node row
    int  d4 = (int)(i4 & 15);          // float4 column within row
    long f  = i4 * 4;                  // flat float index
    v4f e = (n < NUM_USERS)
                ? *(const v4f*)(user_emb + f)
                : *(const v4f*)(item_emb + (f - (long)NUM_USERS * DIM));
    *(v4f*)(emb + f)  = e;
    *(v4f*)(side + f) = (v4f){0.f, 0.f, 0.f, 0.f};
    *(v4f*)(out + n * OUT_COLS + d4 * 4) = e;
}

// ---------------------------------------------------------------------------
// Kernel 2: COO SpMM  side[r,:] += val * emb[c,:]
// One wave (32 lanes) per edge; each lane handles 2 consecutive dims
// (float2 gather + two global_atomic_add_f32, no return). side/emb working
// set (153.6 MB) fits in the 192 MB L2, so the random scatter stays on-chip.
// ---------------------------------------------------------------------------
__global__ void __launch_bounds__(256) ngcf_spmm(
    const int* __restrict__ rows, const int* __restrict__ cols,
    const float* __restrict__ vals, const float* __restrict__ emb,
    float* __restrict__ side, int nnz)
{
    int wave = (int)((blockIdx.x * (long)blockDim.x + threadIdx.x) >> 5);
    int lane = threadIdx.x & 31;
    if (wave >= nnz) return;
    int   r = rows[wave];           // wave-uniform (broadcast load)
    int   c = cols[wave];
    float v = vals[wave];
    const float* src = emb  + (long)c * DIM + 2 * lane;
    float*       dst = side + (long)r * DIM + 2 * lane;
    v2f x = *(const v2f*)src;
    atomicAdd(dst + 0, v * x.x);
    atomicAdd(dst + 1, v * x.y);
}

// ---------------------------------------------------------------------------
// Kernel 3: per-layer dense update, one wave per 16-node tile.
//   out16x64 = S @ Wc + bc + (S*E) @ We + be ; leaky_relu(0.2)
//   emb <- out16x64 (in place); out[:, col0:col0+64] <- row-normalized copy
// Uses V_WMMA_F32_16X16X4_F32: 4 N-chunks x 16 K-steps x 2 matrices = 128 WMMA.
// Also zeroes `side` back to 0 for the next layer / next graph replay.
// Tile traffic is 128-bit per lane (b128 loads/stores).
// ---------------------------------------------------------------------------
__global__ void __launch_bounds__(32) ngcf_dense(
    float* __restrict__ side,                 // read tile, then reset to 0
    float* __restrict__ emb,                  // in/out (in-place per tile)
    const float* __restrict__ Wc, const float* __restrict__ We,   // [64][64] (d_in, d_out)
    const float* __restrict__ bc, const float* __restrict__ be,   // [64]
    float* __restrict__ out, int out_col0)
{
    __shared__ __align__(16) float lS[TILE * DIM];   // side tile, later reused as output tile
    __shared__ __align__(16) float lH[TILE * DIM];   // side * emb tile
    __shared__ float lninv[TILE];

    const int  lane = threadIdx.x;          // 0..31
    const long base = (long)blockIdx.x * TILE;
    const long gbase = base * DIM;

    // Cooperative tile load (16x64 = 256 float4, 8 per lane), zero side behind us.
#pragma unroll
    for (int i = 0; i < 8; ++i) {
        int  q = i * 32 + lane;             // float4 index within tile
        long g = gbase + (long)q * 4;
        v4f s = *(const v4f*)(side + g);
        v4f e = *(const v4f*)(emb + g);
        *(v4f*)&lS[q * 4] = s;
        *(v4f*)&lH[q * 4] = s * e;
        *(v4f*)(side + g) = (v4f){0.f, 0.f, 0.f, 0.f};
    }
    __syncthreads();

    const int hi = lane >> 4;               // 0 or 1
    const int lo = lane & 15;

    // Accumulators: 4 chunks of 16 output columns, each a 16x16 f32 D-matrix.
    v8f acc[4] = {};

    for (int k = 0; k < 16; ++k) {
        // A fragment (16x4 f32 layout): lanes 0-15 hold K=4k,4k+1 of row `lo`,
        // lanes 16-31 hold K=4k+2,4k+3 of row `lo`.
        const int coff = 4 * k + 2 * hi;
        v2f aS = *(const v2f*)&lS[lo * DIM + coff];
        v2f aH = *(const v2f*)&lH[lo * DIM + coff];
#pragma unroll
        for (int j = 0; j < 4; ++j) {
            const int c = 16 * j + lo;      // output column within [0,64)
            // B fragment (4x16 f32): this lane holds rows coff, coff+1 at column c.
            v2f bC; bC.x = Wc[(coff + 0) * DIM + c]; bC.y = Wc[(coff + 1) * DIM + c];
            v2f bE; bE.x = We[(coff + 0) * DIM + c]; bE.y = We[(coff + 1) * DIM + c];
            acc[j] = __builtin_amdgcn_wmma_f32_16x16x4_f32(
                false, aS, false, bC, (short)0, acc[j], false, false);
            acc[j] = __builtin_amdgcn_wmma_f32_16x16x4_f32(
                false, aH, false, bE, (short)0, acc[j], false, false);
        }
    }
    __syncthreads();

    // Bias + leaky_relu(0.2); scatter D-layout back into LDS (reuse lS):
    // lane holds column N = 16j+lo for rows M = v + 8*hi, v in [0,8).
#pragma unroll
    for (int j = 0; j < 4; ++j) {
        const int n = 16 * j + lo;
        const float bias = bc[n] + be[n];
#pragma unroll
        for (int v = 0; v < 8; ++v) {
            const int m = v + 8 * hi;
            float x = acc[j][v] + bias;
            x = (x >= 0.0f) ? x : 0.2f * x;
            lS[m * DIM + n] = x;
        }
    }
    __syncthreads();

    // Row L2 norms (lanes 0-15, staggered column order to avoid bank conflicts).
    if (lane < 16) {
        float ss = 0.0f;
        for (int c0 = 0; c0 < DIM; ++c0) {
            int c = (c0 + lane) & (DIM - 1);
            float x = lS[lane * DIM + c];
            ss += x * x;
        }
        float nrm = sqrtf(ss);
        lninv[lane] = 1.0f / fmaxf(nrm, 1e-12f);
    }
    __syncthreads();

    // Write back (float4): un-normalized -> emb, normalized -> out.
#pragma unroll
    for (int i = 0; i < 8; ++i) {
        int  q = i * 32 + lane;             // float4 index within tile
        int  r = q >> 4;                    // row (16 float4s per row)
        int  c = (q & 15) * 4;              // column
        v4f x = *(const v4f*)&lS[q * 4];
        long node = base + r;
        *(v4f*)(emb + node * DIM + c) = x;
        float ni = lninv[r];
        v4f xn = {x.x * ni, x.y * ni, x.z * ni, x.w * ni};
        *(v4f*)(out + node * OUT_COLS + out_col0 + c) = xn;
    }
}

// ---------------------------------------------------------------------------
extern "C" void kernel_launch(void* const* d_in, const int* in_sizes, int n_in,
                              void* d_out, int out_size, void* d_ws, size_t ws_size,
                              hipStream_t stream)
{
    const int*   rows = (const int*)  d_in[0];
    const int*   cols = (const int*)  d_in[1];
    const float* vals = (const float*)d_in[2];
    const float* uemb = (const float*)d_in[3];
    const float* iemb = (const float*)d_in[4];
    const float* Wc   = (const float*)d_in[5];   // [3,64,64]
    const float* bc   = (const float*)d_in[6];   // [3,64]
    const float* We   = (const float*)d_in[7];   // [3,64,64]
    const float* be   = (const float*)d_in[8];   // [3,64]
    float* out = (float*)d_out;
    const int nnz = in_sizes[0];

    float* side = (float*)d_ws;                          // 300000*64 f32 = 76.8 MB
    float* emb  = side + (size_t)N_NODES * DIM;          // 76.8 MB more

    {   // build emb, zero side, write out[:,0:64]
        long total4 = (long)N_NODES * (DIM / 4);
        int  blocks = (int)((total4 + 255) / 256);
        ngcf_init<<<blocks, 256, 0, stream>>>(uemb, iemb, emb, side, out);
    }

    for (int l = 0; l < 3; ++l) {
        int spmm_blocks = (nnz + 7) / 8;                 // 8 waves (edges) per block
        ngcf_spmm<<<spmm_blocks, 256, 0, stream>>>(rows, cols, vals, emb, side, nnz);
        ngcf_dense<<<N_NODES / TILE, 32, 0, stream>>>(
            side, emb,
            Wc + (size_t)l * DIM * DIM, We + (size_t)l * DIM * DIM,
            bc + (size_t)l * DIM,       be + (size_t)l * DIM,
            out, 64 * (l + 1));
    }
}